// GCNEncoder_84421877170205
// MI455X (gfx1250) — compile-verified
//
#include <hip/hip_runtime.h>

typedef __attribute__((ext_vector_type(2))) float v2f;
typedef __attribute__((ext_vector_type(8))) float v8f;

#define DDIM 128

// ---------------------------------------------------------------------------
// GEMM: H[M,128] = X[M,128] @ W[128,128], fp32 via V_WMMA_F32_16X16X4_F32.
// 256 threads = 8 waves; each wave owns a 16-row stripe and produces the full
// 16x128 output stripe (8 tiles of 16x16, K swept in 32 steps of 4).
// W is staged in LDS pre-swizzled into the exact B-fragment layout so each
// fragment is ONE conflict-free ds_load_b64 with an immediate offset.
// ---------------------------------------------------------------------------
__global__ __launch_bounds__(256, 1) void gemm_wmma_f32(const float* __restrict__ X,
                                                        const float* __restrict__ W,
                                                        float* __restrict__ H, int M) {
  // lds_b[(kk*8 + t)*64 + lane*2 + w] = W[4kk + 2*(lane>=16) + w, 16t + (lane&15)]
  __shared__ float lds_b[DDIM * DDIM];  // 64 KB
  const int tid = threadIdx.x;

  for (int i = tid; i < DDIM * DDIM / 4; i += 256) {
    const int r  = i >> 5;          // W row
    const int c0 = (i & 31) << 2;   // starting col (all 4 cols share one t)
    const float4 w = ((const float4*)W)[i];
    const int kk   = r >> 2;
    const int half = (r & 2) << 3;  // lane-group offset: 0 or 16
    const int wsel = r & 1;         // which word of the lane's b64 pair
    const float vals[4] = {w.x, w.y, w.z, w.w};
#pragma unroll
    for (int j = 0; j < 4; ++j) {
      const int c = c0 + j;
      lds_b[((kk << 3) + (c >> 4)) * 64 + ((half + (c & 15)) << 1) + wsel] = vals[j];
    }
  }
  __syncthreads();

  const int wave = tid >> 5;
  const int lane = tid & 31;
  const int lrow = lane & 15;
  const int kh   = (lane >> 4) << 1;            // 0 for lanes 0-15, 2 for 16-31
  const int row0 = blockIdx.x * 128 + wave * 16;
  const int row  = row0 + lrow;
  const int srow = row < M ? row : 0;           // clamp; bad rows masked at store
  const float* xp = X + (size_t)srow * DDIM + kh;
  const float* bp = lds_b + lane * 2;

  v8f c[8];
#pragma unroll
  for (int t = 0; t < 8; ++t) c[t] = (v8f)(0.0f);

  // K loop in 4 chunks of 8 steps: only 8 A fragments (16 VGPRs) live at once.
  for (int ko = 0; ko < 4; ++ko) {
    v2f a[8];
#pragma unroll
    for (int i = 0; i < 8; ++i)
      a[i] = *(const v2f*)(xp + ((ko << 3) + i) * 4);
#pragma unroll
    for (int i = 0; i < 8; ++i) {
      const int kk = (ko << 3) + i;
#pragma unroll
      for (int t = 0; t < 8; ++t) {
        const v2f b = *(const v2f*)(bp + ((kk << 3) + t) * 64);
        c[t] = __builtin_amdgcn_wmma_f32_16x16x4_f32(
            false, a[i], false, b, (short)0, c[t], false, false);
      }
    }
  }

  // D layout: c[t][v] belongs to row0 + v + (lane>=16 ? 8 : 0), col 16t+lrow
  const int rbase = row0 + ((lane >> 4) << 3);
#pragma unroll
  for (int v = 0; v < 8; ++v) {
    const int r = rbase + v;
    if (r < M) {
      float* hp = H + (size_t)r * DDIM + lrow;
#pragma unroll
      for (int t = 0; t < 8; ++t) hp[16 * t] = c[t][v];
    }
  }
}

// ---------------------------------------------------------------------------
// Degree / normalization
// ---------------------------------------------------------------------------
__global__ void deg_init(float* deg, int n) {
  int i = blockIdx.x * 256 + threadIdx.x;
  if (i < n) deg[i] = 1.0f;  // self-loop contribution
}

__global__ void deg_count(const int* __restrict__ dst, float* deg, int E) {
  int e = blockIdx.x * 256 + threadIdx.x;
  if (e < E) unsafeAtomicAdd(&deg[dst[e]], 1.0f);
}

__global__ void deg_rsqrt(float* deg, int n) {
  int i = blockIdx.x * 256 + threadIdx.x;
  if (i < n) deg[i] = rsqrtf(deg[i]);  // deg >= 1 always
}

// ---------------------------------------------------------------------------
// Edge scatter: acc[dst] += dinv[src]*dinv[dst] * h[src]; one wave per edge,
// 32 lanes x float4 = 128 features. f32 global atomic adds (no-return).
// ---------------------------------------------------------------------------
__global__ __launch_bounds__(256) void scatter_edges(const int* __restrict__ src,
                                                     const int* __restrict__ dst,
                                                     const float* __restrict__ dinv,
                                                     const float* __restrict__ h,
                                                     float* __restrict__ acc, int E) {
  const int e = blockIdx.x * 8 + (threadIdx.x >> 5);
  if (e >= E) return;
  const int lane = threadIdx.x & 31;
  const int s = src[e];
  const int d = dst[e];
  const float norm = dinv[s] * dinv[d];
  const float4 v = ((const float4*)(h + (size_t)s * DDIM))[lane];
  float* ap = acc + (size_t)d * DDIM + lane * 4;
  unsafeAtomicAdd(ap + 0, norm * v.x);
  unsafeAtomicAdd(ap + 1, norm * v.y);
  unsafeAtomicAdd(ap + 2, norm * v.z);
  unsafeAtomicAdd(ap + 3, norm * v.w);
}

// ---------------------------------------------------------------------------
// Finalize: h[i,c] = relu(acc[i,c] + dinv[i]^2 * h[i,c] + bias[c])  (in place)
// ---------------------------------------------------------------------------
__global__ __launch_bounds__(256) void finalize_relu(const float* __restrict__ acc,
                                                     float* __restrict__ h,
                                                     const float* __restrict__ dinv,
                                                     const float* __restrict__ bias,
                                                     int total) {
  int idx = blockIdx.x * 256 + threadIdx.x;
  if (idx >= total) return;
  const int node = idx >> 7;
  const int c = idx & 127;
  const float di = dinv[node];
  const float v = acc[idx] + di * di * h[idx] + bias[c];
  h[idx] = v > 0.0f ? v : 0.0f;
}

// ---------------------------------------------------------------------------
extern "C" void kernel_launch(void* const* d_in, const int* in_sizes, int n_in,
                              void* d_out, int out_size, void* d_ws, size_t ws_size,
                              hipStream_t stream) {
  const float* x  = (const float*)d_in[0];
  const int*   ei = (const int*)d_in[1];   // edge_index, [2, E] row-major
  const float* W1 = (const float*)d_in[2];
  const float* b1 = (const float*)d_in[3];
  const float* W2 = (const float*)d_in[4];
  const float* b2 = (const float*)d_in[5];

  const int n = in_sizes[0] / DDIM;        // 100000 nodes
  const int E = in_sizes[1] / 2;           // 1,600,000 edges
  const int* src = ei;
  const int* dst = ei + E;

  // workspace layout: dinv | A (n*128) | B (n*128)
  float* dinv = (float*)d_ws;
  size_t off = (((size_t)n * 4 + 255) / 256) * 256;
  float* A = (float*)((char*)d_ws + off);
  float* B = A + (size_t)n * DDIM;
  float* out = (float*)d_out;

  const size_t featBytes = (size_t)n * DDIM * sizeof(float);
  const int total = n * DDIM;

  // normalization: deg = 1 + indeg; dinv = rsqrt(deg)
  deg_init <<<(n + 255) / 256, 256, 0, stream>>>(dinv, n);
  deg_count<<<(E + 255) / 256, 256, 0, stream>>>(dst, dinv, E);
  deg_rsqrt<<<(n + 255) / 256, 256, 0, stream>>>(dinv, n);

  // ---- layer 1: h1 = x@W1 -> A; neighbor sum into d_out; bias+relu into A
  gemm_wmma_f32<<<(n + 127) / 128, 256, 0, stream>>>(x, W1, A, n);
  hipMemsetAsync(out, 0, featBytes, stream);
  scatter_edges<<<(E + 7) / 8, 256, 0, stream>>>(src, dst, dinv, A, out, E);
  finalize_relu<<<(total + 255) / 256, 256, 0, stream>>>(out, A, dinv, b1, total);

  // ---- layer 2: h2 = A@W2 -> d_out; neighbor sum into B; bias+relu into d_out
  gemm_wmma_f32<<<(n + 127) / 128, 256, 0, stream>>>(A, W2, out, n);
  hipMemsetAsync(B, 0, featBytes, stream);
  scatter_edges<<<(E + 7) / 8, 256, 0, stream>>>(src, dst, dinv, out, B, E);
  finalize_relu<<<(total + 255) / 256, 256, 0, stream>>>(B, out, dinv, b2, total);
}